// EvolveGCNHModel_50483045597458
// MI455X (gfx1250) — compile-verified
//
#include <hip/hip_runtime.h>
#include <hip/hip_bf16.h>

typedef __attribute__((ext_vector_type(2))) float v2f;
typedef __attribute__((ext_vector_type(8))) float v8f;

#if defined(__has_builtin)
#if __has_builtin(__builtin_amdgcn_wmma_f32_16x16x4_f32)
#define HAVE_WMMA_F32 1
#endif
#endif
#ifndef HAVE_WMMA_F32
#define HAVE_WMMA_F32 0
#endif

#define DIM 64

// ---------------------------------------------------------------------------
// K1: raw projection scores s[i] = dot(x[i], p)  (norm applied later; ordering
// for top-k is invariant under positive scaling). One wave per node.
// ---------------------------------------------------------------------------
__global__ __launch_bounds__(256) void score_kernel(const float* __restrict__ x,
                                                    const float* __restrict__ p,
                                                    float* __restrict__ score, int nN) {
    int wave = blockIdx.x * (blockDim.x >> 5) + (threadIdx.x >> 5);
    int lane = threadIdx.x & 31;
    if (wave >= nN) return;
    const float* row = x + (long)wave * DIM;
    float s = row[lane] * p[lane] + row[lane + 32] * p[lane + 32];
    for (int o = 16; o > 0; o >>= 1) s += __shfl_down(s, o);
    if (lane == 0) score[wave] = s;
}

// ---------------------------------------------------------------------------
// K2: exact top-64 by 64 argmax sweeps (single block, lowest index wins ties,
// matching jax.lax.top_k). Mutates the ws score copy only.
// ---------------------------------------------------------------------------
__global__ __launch_bounds__(1024) void topk_kernel(float* __restrict__ score,
                                                    int* __restrict__ perm,
                                                    float* __restrict__ vals, int nN) {
    __shared__ float sv[1024];
    __shared__ int   si[1024];
    int tid = threadIdx.x;
    for (int it = 0; it < DIM; ++it) {
        float best = -__builtin_huge_valf();
        int   bidx = 0x7fffffff;
        for (int i = tid; i < nN; i += 1024) {
            float v = score[i];
            if (v > best || (v == best && i < bidx)) { best = v; bidx = i; }
        }
        sv[tid] = best; si[tid] = bidx;
        __syncthreads();
        for (int s = 512; s > 0; s >>= 1) {
            if (tid < s) {
                if (sv[tid + s] > sv[tid] ||
                    (sv[tid + s] == sv[tid] && si[tid + s] < si[tid])) {
                    sv[tid] = sv[tid + s]; si[tid] = si[tid + s];
                }
            }
            __syncthreads();
        }
        if (tid == 0) {
            perm[it] = si[0];
            vals[it] = sv[0];
            score[si[0]] = -__builtin_huge_valf();
        }
        __syncthreads();
    }
}

// ---------------------------------------------------------------------------
// K3: x_tilde = x[perm] * tanh(vals/||p||), then one GRU step with h0 = init_W
// producing the evolved GCN weight W [64,64]. Tiny one-shot kernel.
// ---------------------------------------------------------------------------
__global__ __launch_bounds__(256) void gru_kernel(const float* __restrict__ x,
                                                  const float* __restrict__ p,
                                                  const int* __restrict__ perm,
                                                  const float* __restrict__ vals,
                                                  const float* __restrict__ h0g,
                                                  const float* __restrict__ Wih,
                                                  const float* __restrict__ Whh,
                                                  const float* __restrict__ bih,
                                                  const float* __restrict__ bhh,
                                                  float* __restrict__ Wout) {
    __shared__ float xt[DIM * DIM];
    __shared__ float h0[DIM * DIM];
    __shared__ float normp;
    int tid = threadIdx.x;
    if (tid == 0) {
        float s = 0.f;
        for (int k = 0; k < DIM; ++k) s += p[k] * p[k];
        normp = sqrtf(s);
    }
    __syncthreads();
    for (int idx = tid; idx < DIM * DIM; idx += 256) {
        int j = idx >> 6, c = idx & 63;
        xt[idx] = x[(long)perm[j] * DIM + c] * tanhf(vals[j] / normp);
        h0[idx] = h0g[idx];
    }
    __syncthreads();
    for (int idx = tid; idx < DIM * DIM; idx += 256) {
        int j = idx >> 6, c = idx & 63;
        float ir = bih[c], iz = bih[DIM + c], in_ = bih[2 * DIM + c];
        float hr = bhh[c], hz = bhh[DIM + c], hn  = bhh[2 * DIM + c];
        for (int k = 0; k < DIM; ++k) {
            float xv = xt[j * DIM + k], hv = h0[j * DIM + k];
            ir  += xv * Wih[c * DIM + k];
            iz  += xv * Wih[(DIM + c) * DIM + k];
            in_ += xv * Wih[(2 * DIM + c) * DIM + k];
            hr  += hv * Whh[c * DIM + k];
            hz  += hv * Whh[(DIM + c) * DIM + k];
            hn  += hv * Whh[(2 * DIM + c) * DIM + k];
        }
        float r  = 1.f / (1.f + expf(-(ir + hr)));
        float z  = 1.f / (1.f + expf(-(iz + hz)));
        float nn = tanhf(in_ + r * hn);
        Wout[idx] = (1.f - z) * nn + z * h0[idx];
    }
}

// ---------------------------------------------------------------------------
// K4: xw = x @ W  [50000x64 @ 64x64], fp32 WMMA 16x16x4. One 16x64 output
// tile per wave (4 accumulators, 16 K-steps). W staged in LDS per block.
// 50000 % 16 == 0, and tile-skip branches are wave-uniform so EXEC stays all-1s.
// ---------------------------------------------------------------------------
__global__ __launch_bounds__(256) void gemm_xw_kernel(const float* __restrict__ x,
                                                      const float* __restrict__ Wev,
                                                      float* __restrict__ xw, int nN) {
    __shared__ float sW[DIM * DIM];
    for (int i = threadIdx.x; i < DIM * DIM; i += 256) sW[i] = Wev[i];
    __syncthreads();
    int wave = threadIdx.x >> 5;
    int lane = threadIdx.x & 31;
    long row0 = ((long)blockIdx.x * 8 + wave) * 16;
    if (row0 + 16 > nN) return;               // wave-uniform
    int m  = lane & 15;                       // M row (A) / N col (B,C)
    int kh = lane >> 4;                       // K half-select per ISA layout
#if HAVE_WMMA_F32
    v8f acc[4] = {};
    const float* arow = x + (row0 + m) * DIM;
    for (int kt = 0; kt < 16; ++kt) {
        int k = kt * 4 + kh * 2;
        v2f a; a.x = arow[k]; a.y = arow[k + 1];
#pragma unroll
        for (int nt = 0; nt < 4; ++nt) {
            v2f b;
            b.x = sW[k * DIM + nt * 16 + m];
            b.y = sW[(k + 1) * DIM + nt * 16 + m];
            acc[nt] = __builtin_amdgcn_wmma_f32_16x16x4_f32(
                false, a, false, b, (short)0, acc[nt], false, false);
        }
    }
#pragma unroll
    for (int nt = 0; nt < 4; ++nt)
#pragma unroll
        for (int v = 0; v < 8; ++v)
            xw[(row0 + v + 8 * kh) * DIM + nt * 16 + m] = acc[nt][v];
#else
    for (int r = 0; r < 16; ++r) {
        const float* arow = x + (row0 + r) * DIM;
        float a0 = 0.f, a1 = 0.f;
        for (int k = 0; k < DIM; ++k) {
            float xv = arow[k];
            a0 += xv * sW[k * DIM + lane];
            a1 += xv * sW[k * DIM + lane + 32];
        }
        xw[(row0 + r) * DIM + lane]      = a0;
        xw[(row0 + r) * DIM + lane + 32] = a1;
    }
#endif
}

// ---------------------------------------------------------------------------
// K5: degree accumulation over targets (+1 self-loop per node).
// ---------------------------------------------------------------------------
__global__ __launch_bounds__(256) void deg_kernel(const int* __restrict__ col,
                                                  const float* __restrict__ ew,
                                                  float* __restrict__ deg,
                                                  int nE, int nN) {
    int t = blockIdx.x * blockDim.x + threadIdx.x;
    if (t < nE)            atomicAdd(&deg[col[t]], ew[t]);
    else if (t < nE + nN)  atomicAdd(&deg[t - nE], 1.0f);
}

// K6: dinv = deg > 0 ? rsqrt(deg) : 0   (in place)
__global__ __launch_bounds__(256) void dinv_kernel(float* __restrict__ deg, int nN) {
    int i = blockIdx.x * blockDim.x + threadIdx.x;
    if (i < nN) { float d = deg[i]; deg[i] = d > 0.f ? rsqrtf(d) : 0.f; }
}

// ---------------------------------------------------------------------------
// K7: edge scatter  agg[col] += dinv[row]*w*dinv[col] * xw[row].
// One wave per edge, 2 channels per lane; f32 atomics land in L2 (both
// 12.8 MB arrays fit in the 192 MB L2).
// ---------------------------------------------------------------------------
__global__ __launch_bounds__(256) void scatter_kernel(const int* __restrict__ ei,
                                                      const float* __restrict__ ew,
                                                      const float* __restrict__ dinv,
                                                      const float* __restrict__ xw,
                                                      float* __restrict__ agg, int nE) {
    int wave = blockIdx.x * (blockDim.x >> 5) + (threadIdx.x >> 5);
    int lane = threadIdx.x & 31;
    if (wave >= nE) return;
    int r = ei[wave], c = ei[nE + wave];
    float nrm = dinv[r] * ew[wave] * dinv[c];
    const float* src = xw + (long)r * DIM;
    float* dst = agg + (long)c * DIM;
    atomicAdd(&dst[lane],      nrm * src[lane]);
    atomicAdd(&dst[lane + 32], nrm * src[lane + 32]);
}

// ---------------------------------------------------------------------------
// K8: head. Self-loop folded in analytically: h = relu(agg + dinv^2 * xw),
// then out = h . lin_W + lin_b. Wave per node, shuffle reduction.
// ---------------------------------------------------------------------------
__global__ __launch_bounds__(256) void head_kernel(const float* __restrict__ agg,
                                                   const float* __restrict__ dinv,
                                                   const float* __restrict__ xw,
                                                   const float* __restrict__ linW,
                                                   const float* __restrict__ linb,
                                                   float* __restrict__ out, int nN) {
    int wave = blockIdx.x * (blockDim.x >> 5) + (threadIdx.x >> 5);
    int lane = threadIdx.x & 31;
    if (wave >= nN) return;
    float di = dinv[wave], sl = di * di;
    long base = (long)wave * DIM;
    float a0 = agg[base + lane]      + sl * xw[base + lane];
    float a1 = agg[base + lane + 32] + sl * xw[base + lane + 32];
    float s = fmaxf(a0, 0.f) * linW[lane] + fmaxf(a1, 0.f) * linW[lane + 32];
    for (int o = 16; o > 0; o >>= 1) s += __shfl_down(s, o);
    if (lane == 0) out[wave] = s + linb[0];
}

extern "C" void kernel_launch(void* const* d_in, const int* in_sizes, int n_in,
                              void* d_out, int out_size, void* d_ws, size_t ws_size,
                              hipStream_t stream) {
    const float* x      = (const float*)d_in[0];
    const int*   ei     = (const int*)  d_in[1];   // [2, E] : row then col
    const float* ew     = (const float*)d_in[2];
    const float* pool_p = (const float*)d_in[3];
    const float* init_W = (const float*)d_in[4];
    const float* Wih    = (const float*)d_in[5];
    const float* Whh    = (const float*)d_in[6];
    const float* bih    = (const float*)d_in[7];
    const float* bhh    = (const float*)d_in[8];
    const float* linW   = (const float*)d_in[9];
    const float* linb   = (const float*)d_in[10];
    float*       out    = (float*)d_out;

    const int nN = in_sizes[0] / DIM;   // 50000
    const int nE = in_sizes[2];         // 800000

    // ---- workspace layout (all offsets 512B-aligned) ----
    char* ws = (char*)d_ws;
    size_t off = 0;
    float* score = (float*)(ws + off); off += ((size_t)nN * 4 + 511) & ~511ull;
    int*   perm  = (int*)  (ws + off); off += 512;
    float* vals  = (float*)(ws + off); off += 512;
    float* Wev   = (float*)(ws + off); off += DIM * DIM * 4;
    float* deg   = (float*)(ws + off); off += ((size_t)nN * 4 + 511) & ~511ull;
    float* xw    = (float*)(ws + off); off += (size_t)nN * DIM * 4;
    float* agg   = (float*)(ws + off); off += (size_t)nN * DIM * 4;

    hipMemsetAsync(deg, 0, (size_t)nN * sizeof(float), stream);
    hipMemsetAsync(agg, 0, (size_t)nN * DIM * sizeof(float), stream);

    score_kernel<<<(nN + 7) / 8, 256, 0, stream>>>(x, pool_p, score, nN);
    topk_kernel<<<1, 1024, 0, stream>>>(score, perm, vals, nN);
    gru_kernel<<<1, 256, 0, stream>>>(x, pool_p, perm, vals, init_W,
                                      Wih, Whh, bih, bhh, Wev);
    int mtiles = nN / 16;               // 3125 (exact)
    gemm_xw_kernel<<<(mtiles + 7) / 8, 256, 0, stream>>>(x, Wev, xw, nN);
    deg_kernel<<<(nE + nN + 255) / 256, 256, 0, stream>>>(ei + nE, ew, deg, nE, nN);
    dinv_kernel<<<(nN + 255) / 256, 256, 0, stream>>>(deg, nN);
    scatter_kernel<<<(nE + 7) / 8, 256, 0, stream>>>(ei, ew, deg, xw, agg, nE);
    head_kernel<<<(nN + 7) / 8, 256, 0, stream>>>(agg, deg, xw, linW, linb, out, nN);
}